// ClassifierV4_46231027974388
// MI455X (gfx1250) — compile-verified
//
#include <hip/hip_runtime.h>
#include <hip/hip_bf16.h>

#define N_NODES     100000   // == 6250 * 16, so 16-row M tiles divide exactly
#define N_EDGES     1200000
#define NUM_FEATS   256
#define DIM         64
#define NUM_CLASSES 40
#define NUM_LAYERS  8

// h_new = 0.8*agg + 0.1*h + 0.1*h0  (1 - RESIDUAL_WEIGHT - BETA = 0.8)
#define W_AGG 0.8f
#define W_RES 0.1f
#define W_ORI 0.1f

typedef __attribute__((ext_vector_type(2))) float v2f;
typedef __attribute__((ext_vector_type(8))) float v8f;

__device__ __forceinline__ v8f wmma4(v2f a, v2f b, v8f c) {
  // V_WMMA_F32_16X16X4_F32 : D = A(16x4) * B(4x16) + C(16x16), fp32 throughout
  return __builtin_amdgcn_wmma_f32_16x16x4_f32(
      /*neg_a=*/false, a, /*neg_b=*/false, b,
      /*c_mod=*/(short)0, c, /*reuse_a=*/false, /*reuse_b=*/false);
}

// ---------------------------------------------------------------- utilities
__global__ void zero_f32(float* __restrict__ p, long n) {
  long i = (long)blockIdx.x * blockDim.x + threadIdx.x;
  long stride = (long)gridDim.x * blockDim.x;
  for (; i < n; i += stride) p[i] = 0.0f;
}

__global__ void degree_kernel(const int* __restrict__ ei,
                              const float* __restrict__ ew,
                              float* __restrict__ deg) {
  int e = blockIdx.x * blockDim.x + threadIdx.x;
  if (e < N_EDGES) atomicAdd(&deg[ei[e]], ew[e]);  // row = ei[0..E-1]
}

__global__ void dinv_kernel(float* __restrict__ deg) {
  int i = blockIdx.x * blockDim.x + threadIdx.x;
  if (i < N_NODES) {
    float d = deg[i];
    deg[i] = (d > 0.0f) ? rsqrtf(d) : 0.0f;
  }
}

__global__ void norm_kernel(const int* __restrict__ ei,
                            const float* __restrict__ ew,
                            const float* __restrict__ dinv,
                            float* __restrict__ nrm) {
  int e = blockIdx.x * blockDim.x + threadIdx.x;
  if (e < N_EDGES) {
    int r = ei[e];
    int c = ei[N_EDGES + e];
    nrm[e] = dinv[r] * ew[e] * dinv[c];
  }
}

// ------------------------------------------------ input GEMM: h0 = h = relu(x@W_in+b)
// One wave computes a 16-row stripe across all 64 output columns (4 N-tiles).
// Also zero-fills agg for the first scatter pass (fused, same store footprint).
__global__ void __launch_bounds__(256) input_gemm_relu(
    const float* __restrict__ X, const float* __restrict__ W,
    const float* __restrict__ bias, float* __restrict__ h0,
    float* __restrict__ h, float* __restrict__ agg) {
  const int wave = threadIdx.x >> 5;
  const int lane = threadIdx.x & 31;
  const int tile = blockIdx.x * 8 + wave;
  const int m0 = tile * 16;
  if (m0 >= N_NODES) return;            // wave-uniform; EXEC stays all-ones for WMMA
  const int hi  = lane >> 4;            // half-wave select
  const int l16 = lane & 15;
  const int arow = m0 + l16;            // always < N_NODES (tiles divide exactly)

  v8f zero = {};
  v8f acc[4];
#pragma unroll
  for (int t = 0; t < 4; ++t) acc[t] = zero;

  for (int k0 = 0; k0 < NUM_FEATS; k0 += 4) {
    const int kk = k0 + hi * 2;         // A layout: v0 = K0|K2, v1 = K1|K3
    v2f a;
    a.x = X[(long)arow * NUM_FEATS + kk];
    a.y = X[(long)arow * NUM_FEATS + kk + 1];
#pragma unroll
    for (int t = 0; t < 4; ++t) {
      v2f b;                            // B layout mirrors A across lane halves
      b.x = W[(long)kk * DIM + t * 16 + l16];
      b.y = W[(long)(kk + 1) * DIM + t * 16 + l16];
      acc[t] = wmma4(a, b, acc[t]);
    }
  }

#pragma unroll
  for (int t = 0; t < 4; ++t) {
    const int col = t * 16 + l16;
    const float bv = bias[col];
#pragma unroll
    for (int r = 0; r < 8; ++r) {
      const long idx = (long)(m0 + r + hi * 8) * DIM + col;  // C/D: vgpr r -> M=r|r+8
      float v = acc[t][r] + bv;
      v = v > 0.0f ? v : 0.0f;
      h0[idx]  = v;
      h [idx]  = v;
      agg[idx] = 0.0f;
    }
  }
}

// ------------------------------------------------ edge scatter: agg[dst] += norm*h[src]
__global__ void __launch_bounds__(256) scatter_kernel(
    const int* __restrict__ ei, const float* __restrict__ nrm,
    const float* __restrict__ h, float* __restrict__ agg) {
  long idx = (long)blockIdx.x * 256 + threadIdx.x;
  long e = idx >> 6;
  int  d = (int)(idx & 63);
  if (e >= N_EDGES) return;
  int src = ei[e];
  int dst = ei[N_EDGES + e];
  atomicAdd(&agg[(long)dst * DIM + d], nrm[e] * h[(long)src * DIM + d]);
}

// ---------- layer: h = relu((0.8*agg + 0.1*h + 0.1*h0) @ W_l), in place on h.
// Safe: each wave owns its 16-row stripe; all reads of those rows complete
// before its stores. Also re-zeroes agg for the next layer's scatter (fused).
__global__ void __launch_bounds__(256) layer_gemm_relu(
    float* __restrict__ agg, float* __restrict__ h,
    const float* __restrict__ h0, const float* __restrict__ Wl) {
  const int wave = threadIdx.x >> 5;
  const int lane = threadIdx.x & 31;
  const int tile = blockIdx.x * 8 + wave;
  const int m0 = tile * 16;
  if (m0 >= N_NODES) return;
  const int hi  = lane >> 4;
  const int l16 = lane & 15;
  const int arow = m0 + l16;

  v8f zero = {};
  v8f acc[4];
#pragma unroll
  for (int t = 0; t < 4; ++t) acc[t] = zero;

  for (int k0 = 0; k0 < DIM; k0 += 4) {
    const int kk = k0 + hi * 2;
    const long i0 = (long)arow * DIM + kk;
    v2f a;
    a.x = W_AGG * agg[i0]     + W_RES * h[i0]     + W_ORI * h0[i0];
    a.y = W_AGG * agg[i0 + 1] + W_RES * h[i0 + 1] + W_ORI * h0[i0 + 1];
#pragma unroll
    for (int t = 0; t < 4; ++t) {
      v2f b;
      b.x = Wl[(long)kk * DIM + t * 16 + l16];
      b.y = Wl[(long)(kk + 1) * DIM + t * 16 + l16];
      acc[t] = wmma4(a, b, acc[t]);
    }
  }

#pragma unroll
  for (int t = 0; t < 4; ++t) {
    const int col = t * 16 + l16;
#pragma unroll
    for (int r = 0; r < 8; ++r) {
      const long idx = (long)(m0 + r + hi * 8) * DIM + col;
      float v = acc[t][r];
      h[idx]   = v > 0.0f ? v : 0.0f;
      agg[idx] = 0.0f;
    }
  }
}

// ------------------------------------------- output GEMM: y = h @ W_out + b_out
__global__ void __launch_bounds__(256) out_gemm(
    const float* __restrict__ h, const float* __restrict__ W,
    const float* __restrict__ bias, float* __restrict__ y) {
  const int wave = threadIdx.x >> 5;
  const int lane = threadIdx.x & 31;
  const int tile = blockIdx.x * 8 + wave;
  const int m0 = tile * 16;
  if (m0 >= N_NODES) return;
  const int hi  = lane >> 4;
  const int l16 = lane & 15;
  const int arow = m0 + l16;
  const int colsafe = 32 + l16 < NUM_CLASSES ? 32 + l16 : NUM_CLASSES - 1;

  v8f zero = {};
  v8f acc[3];
#pragma unroll
  for (int t = 0; t < 3; ++t) acc[t] = zero;

  for (int k0 = 0; k0 < DIM; k0 += 4) {
    const int kk = k0 + hi * 2;
    v2f a;
    a.x = h[(long)arow * DIM + kk];
    a.y = h[(long)arow * DIM + kk + 1];
#pragma unroll
    for (int t = 0; t < 3; ++t) {
      v2f b;
      if (t < 2) {                       // cols 0..31: always valid
        const int col = t * 16 + l16;
        b.x = W[(long)kk * NUM_CLASSES + col];
        b.y = W[(long)(kk + 1) * NUM_CLASSES + col];
      } else {                           // cols 32..47: clamp addr, select zero
        const bool ok = (32 + l16) < NUM_CLASSES;
        float bx = W[(long)kk * NUM_CLASSES + colsafe];
        float by = W[(long)(kk + 1) * NUM_CLASSES + colsafe];
        b.x = ok ? bx : 0.0f;
        b.y = ok ? by : 0.0f;
      }
      acc[t] = wmma4(a, b, acc[t]);
    }
  }

#pragma unroll
  for (int t = 0; t < 3; ++t) {
    const int col = t * 16 + l16;
    if (t < 2) {
      const float bv = bias[col];
#pragma unroll
      for (int r = 0; r < 8; ++r)
        y[(long)(m0 + r + hi * 8) * NUM_CLASSES + col] = acc[t][r] + bv;
    } else if (col < NUM_CLASSES) {
      const float bv = bias[col];
#pragma unroll
      for (int r = 0; r < 8; ++r)
        y[(long)(m0 + r + hi * 8) * NUM_CLASSES + col] = acc[t][r] + bv;
    }
  }
}

// ----------------------------------------------------- row-wise log-softmax in place
__global__ void log_softmax_rows(float* __restrict__ y) {
  int row = blockIdx.x * blockDim.x + threadIdx.x;
  if (row >= N_NODES) return;
  float* p = y + (long)row * NUM_CLASSES;
  float m = p[0];
#pragma unroll
  for (int c = 1; c < NUM_CLASSES; ++c) m = fmaxf(m, p[c]);
  float s = 0.0f;
#pragma unroll
  for (int c = 0; c < NUM_CLASSES; ++c) s += expf(p[c] - m);
  float ls = logf(s);
#pragma unroll
  for (int c = 0; c < NUM_CLASSES; ++c) p[c] = p[c] - m - ls;
}

// ---------------------------------- lconstr = GAMMA * sum_l ||W_l - I||_F (GAMMA=1)
__global__ void __launch_bounds__(256) loss_kernel(const float* __restrict__ gw,
                                                   float* __restrict__ out) {
  __shared__ float red[256];
  const int tid = threadIdx.x;
  float total = 0.0f;
  for (int l = 0; l < NUM_LAYERS; ++l) {
    float s = 0.0f;
    for (int i = tid; i < DIM * DIM; i += 256) {
      float tgt = ((i / DIM) == (i % DIM)) ? 1.0f : 0.0f;  // sqrt(C_MAX)=1
      float d = gw[l * DIM * DIM + i] - tgt;
      s += d * d;
    }
    red[tid] = s;
    __syncthreads();
    for (int off = 128; off > 0; off >>= 1) {
      if (tid < off) red[tid] += red[tid + off];
      __syncthreads();
    }
    if (tid == 0) total += sqrtf(red[0]);
    __syncthreads();
  }
  if (tid == 0) out[0] = total;
}

extern "C" void kernel_launch(void* const* d_in, const int* in_sizes, int n_in,
                              void* d_out, int out_size, void* d_ws, size_t ws_size,
                              hipStream_t stream) {
  const float* x     = (const float*)d_in[0];
  const int*   ei    = (const int*)  d_in[1];   // [2, E] flat: rows then cols
  const float* ew    = (const float*)d_in[2];
  const float* W_in  = (const float*)d_in[3];
  const float* b_in  = (const float*)d_in[4];
  const float* gcn_w = (const float*)d_in[5];   // [8, 64, 64]
  const float* W_out = (const float*)d_in[6];
  const float* b_out = (const float*)d_in[7];

  float* y = (float*)d_out;                        // [N, 40] then [1] lconstr
  float* ws   = (float*)d_ws;
  float* deg  = ws;                                // N      (becomes deg_inv_sqrt)
  float* nrm  = deg + N_NODES;                     // E
  float* h0   = nrm + N_EDGES;                     // N*DIM
  float* h    = h0 + (long)N_NODES * DIM;          // N*DIM
  float* agg  = h  + (long)N_NODES * DIM;          // N*DIM

  const int edge_blocks = (N_EDGES + 255) / 256;
  const int node_blocks = (N_NODES + 255) / 256;
  const int gemm_blocks = ((N_NODES / 16) + 7) / 8;           // 6250 tiles / 8 waves
  const long scatter_threads = (long)N_EDGES * DIM;
  const int scatter_blocks = (int)((scatter_threads + 255) / 256);

  // norm = dinv[src] * w * dinv[dst]
  zero_f32<<<node_blocks, 256, 0, stream>>>(deg, (long)N_NODES);
  degree_kernel<<<edge_blocks, 256, 0, stream>>>(ei, ew, deg);
  dinv_kernel<<<node_blocks, 256, 0, stream>>>(deg);
  norm_kernel<<<edge_blocks, 256, 0, stream>>>(ei, ew, deg, nrm);

  // h0 = h = relu(x @ W_in + b_in); agg <- 0 (fused)
  input_gemm_relu<<<gemm_blocks, 256, 0, stream>>>(x, W_in, b_in, h0, h, agg);

  for (int l = 0; l < NUM_LAYERS; ++l) {
    scatter_kernel<<<scatter_blocks, 256, 0, stream>>>(ei, nrm, h, agg);
    layer_gemm_relu<<<gemm_blocks, 256, 0, stream>>>(
        agg, h, h0, gcn_w + (long)l * DIM * DIM);   // also re-zeroes agg
  }

  out_gemm<<<gemm_blocks, 256, 0, stream>>>(h, W_out, b_out, y);
  log_softmax_rows<<<node_blocks, 256, 0, stream>>>(y);
  loss_kernel<<<1, 256, 0, stream>>>(gcn_w, y + (long)N_NODES * NUM_CLASSES);
}